// PINNSimulator_37855841747580
// MI455X (gfx1250) — compile-verified
//
#include <hip/hip_runtime.h>

// ---------------------------------------------------------------------------
// PINN Fisher-Kolmogorov loss, MI455X (gfx1250).
// Memory-bound: 6 x 56.6MB fp32 reads -> ~15us floor @ 23.3 TB/s.
// Strategy: b128 vector loads, NT temporal hints on single-use streams,
// RT (cached) loads for the stencil-reused u tensor (fits in 192MB L2),
// wave32 shuffle reductions, deterministic 2-pass reduce, and a
// v_wmma_f32_16x16x4_f32 ones-matrix reduction step in the tail kernel.
// ---------------------------------------------------------------------------

typedef float v4f __attribute__((ext_vector_type(4)));
typedef float v2f __attribute__((ext_vector_type(2)));
typedef float v8f __attribute__((ext_vector_type(8)));

#define DIM     192
#define NTOT    (2 * DIM * DIM * DIM)       // 14,155,776 elements
#define NV4     (NTOT / 4)                  // 3,538,944 float4 packets
#define ZV      (DIM / 4)                   // 48 float4 per z-line
#define SY      DIM                         // y stride (floats)
#define SX      (DIM * DIM)                 // x stride (floats)

#define D_MIN   0.001f
#define D_RANGE (0.02f - 0.001f)
#define R_MIN   0.012f
#define R_RANGE (0.034f - 0.012f)

__device__ __forceinline__ float sig1(float x) {
    return 1.0f / (1.0f + __expf(-x));
}
__device__ __forceinline__ v4f sig4(v4f x) {
    v4f r;
    r.x = sig1(x.x); r.y = sig1(x.y); r.z = sig1(x.z); r.w = sig1(x.w);
    return r;
}
__device__ __forceinline__ float hsum4(v4f v) {
    return (v.x + v.y) + (v.z + v.w);
}

// ---------------------------------------------------------------------------
// Kernel 1: grid-stride over float4 packets; per-block partial sums -> ws.
// ---------------------------------------------------------------------------
__global__ __launch_bounds__(256)
void pinn_partial_kernel(const float* __restrict__ u,
                         const float* __restrict__ dudt,
                         const float* __restrict__ rawD,
                         const float* __restrict__ rawR,
                         const float* __restrict__ u0,
                         const float* __restrict__ t0,
                         float* __restrict__ ws)
{
    const v4f vzero = {0.f, 0.f, 0.f, 0.f};
    float ic_acc = 0.f;
    float ph_acc = 0.f;

    const int stride = gridDim.x * blockDim.x;
    for (int v = blockIdx.x * blockDim.x + threadIdx.x; v < NV4; v += stride) {
        const int z4 = v % ZV;
        const int t  = v / ZV;
        const int y  = t % DIM;
        const int x  = (t / DIM) % DIM;
        const int base = v * 4;            // element offset (< 2^24, fits int)

        // ---- 7-point Laplacian on u (zero-padded boundaries) -------------
        const v4f uc = *(const v4f*)(u + base);                 // RT: reused 7x, lives in L2
        const v4f ym = (y > 0)       ? *(const v4f*)(u + base - SY) : vzero;
        const v4f yp = (y < DIM - 1) ? *(const v4f*)(u + base + SY) : vzero;
        const v4f xm = (x > 0)       ? *(const v4f*)(u + base - SX) : vzero;
        const v4f xp = (x < DIM - 1) ? *(const v4f*)(u + base + SX) : vzero;
        const float zl = (z4 > 0)      ? u[base - 1] : 0.f;
        const float zr = (z4 < ZV - 1) ? u[base + 4] : 0.f;
        const v4f zm = {zl,   uc.x, uc.y, uc.z};
        const v4f zp = {uc.y, uc.z, uc.w, zr};
        const v4f lap = (ym + yp) + (xm + xp) + (zm + zp) - 6.0f * uc;

        // ---- single-use streams: non-temporal (don't evict u from L2) ----
        const v4f dt = __builtin_nontemporal_load((const v4f*)(dudt + base));
        const v4f rd = __builtin_nontemporal_load((const v4f*)(rawD + base));
        const v4f rr = __builtin_nontemporal_load((const v4f*)(rawR + base));
        const v4f a0 = __builtin_nontemporal_load((const v4f*)(u0   + base));
        const v4f b0 = __builtin_nontemporal_load((const v4f*)(t0   + base));

        const v4f Dm = D_MIN + sig4(rd) * D_RANGE;
        const v4f Rm = R_MIN + sig4(rr) * R_RANGE;

        const v4f res = dt - Dm * lap - Rm * uc * (1.0f - uc);
        ph_acc += hsum4(res * res);

        const v4f dic = a0 - b0;
        ic_acc += hsum4(dic * dic);
    }

    // ---- wave32 shuffle reduction, then cross-wave via LDS ---------------
    for (int off = 16; off > 0; off >>= 1) {
        ic_acc += __shfl_down(ic_acc, off, 32);
        ph_acc += __shfl_down(ph_acc, off, 32);
    }
    __shared__ float sic[8];
    __shared__ float sph[8];
    const int wid  = threadIdx.x >> 5;
    const int lane = threadIdx.x & 31;
    if (lane == 0) { sic[wid] = ic_acc; sph[wid] = ph_acc; }
    __syncthreads();
    if (threadIdx.x == 0) {
        float a = 0.f, b = 0.f;
        for (int i = 0; i < 8; ++i) { a += sic[i]; b += sph[i]; }
        ws[2 * blockIdx.x + 0] = a;
        ws[2 * blockIdx.x + 1] = b;
    }
}

// ---------------------------------------------------------------------------
// Kernel 2: reduce per-block partials, finish with a WMMA ones-matrix
// reduction (D[m][n] = sum_k B[k][n] with A == ones => p[n] + p[n+16]),
// then write [total, loss_ic, loss_physics].
// ---------------------------------------------------------------------------
__global__ __launch_bounds__(256)
void pinn_final_kernel(const float* __restrict__ ws, int nblocks,
                       float* __restrict__ out)
{
    float ic = 0.f, ph = 0.f;
    for (int i = threadIdx.x; i < nblocks; i += 256) {
        ic += ws[2 * i + 0];
        ph += ws[2 * i + 1];
    }
    __shared__ float sic[256];
    __shared__ float sph[256];
    sic[threadIdx.x] = ic;
    sph[threadIdx.x] = ph;
    __syncthreads();

    // Every wave redundantly reduces the 256 partials so EXEC is all-1s
    // for the WMMA (ISA requirement); cost is negligible in this tail.
    const int lane = threadIdx.x & 31;
    float a = 0.f, b = 0.f;
    for (int j = 0; j < 8; ++j) {
        a += sic[lane + 32 * j];
        b += sph[lane + 32 * j];
    }

#if __has_builtin(__builtin_amdgcn_wmma_f32_16x16x4_f32)
    {
        // A (16x4) = all ones -> D[m][n] = B[0][n] + B[2][n].
        // B VGPR0: lanes 0-15 hold row K=0 (N=lane), lanes 16-31 row K=2.
        // So with B = {partial, 0}: D[0] on every lane = p[lane&15] + p[(lane&15)+16].
        const v2f A = {1.0f, 1.0f};
        v8f C = {};
        v2f B = {a, 0.0f};
        v8f D0 = __builtin_amdgcn_wmma_f32_16x16x4_f32(
            false, A, false, B, (short)0, C, false, false);
        a = D0[0];
        B.x = b;
        v8f D1 = __builtin_amdgcn_wmma_f32_16x16x4_f32(
            false, A, false, B, (short)0, C, false, false);
        b = D1[0];
        for (int off = 8; off > 0; off >>= 1) {       // finish 16 -> 1
            a += __shfl_xor(a, off, 32);
            b += __shfl_xor(b, off, 32);
        }
    }
#else
    for (int off = 16; off > 0; off >>= 1) {
        a += __shfl_xor(a, off, 32);
        b += __shfl_xor(b, off, 32);
    }
#endif

    if (threadIdx.x == 0) {
        const float inv_n = 1.0f / (float)NTOT;
        const float loss_ic   = a * inv_n;
        const float loss_phys = b * inv_n;
        out[0] = 10.0f * loss_ic + 1.0f * loss_phys;  // LAMBDA_IC, LAMBDA_PHYS
        out[1] = loss_ic;
        out[2] = loss_phys;
    }
}

// ---------------------------------------------------------------------------
extern "C" void kernel_launch(void* const* d_in, const int* in_sizes, int n_in,
                              void* d_out, int out_size, void* d_ws, size_t ws_size,
                              hipStream_t stream)
{
    const float* u    = (const float*)d_in[0];   // u_pred_t
    const float* dudt = (const float*)d_in[1];
    const float* rawD = (const float*)d_in[2];
    const float* rawR = (const float*)d_in[3];
    const float* u0   = (const float*)d_in[4];   // u_pred_t0
    const float* t0   = (const float*)d_in[5];   // target_t0

    int nblocks = 2048;
    if (ws_size < (size_t)nblocks * 2 * sizeof(float)) {
        nblocks = (int)(ws_size / (2 * sizeof(float)));
        if (nblocks < 1) nblocks = 1;
    }

    pinn_partial_kernel<<<nblocks, 256, 0, stream>>>(
        u, dudt, rawD, rawR, u0, t0, (float*)d_ws);
    pinn_final_kernel<<<1, 256, 0, stream>>>(
        (const float*)d_ws, nblocks, (float*)d_out);
}